// CombinedLoss_16509854286367
// MI455X (gfx1250) — compile-verified
//
#include <hip/hip_runtime.h>
#include <hip/hip_bf16.h>
#include <cstdint>

#define BATCH 8
#define NCLS  19
#define H     512
#define W     512
#define HW    (H * W)
#define NPIX  (BATCH * HW)          // 2,097,152
#define TILE  256                    // pixels per tile (contiguous, same b)
#define NTILES (NPIX / TILE)         // 8192
#define NBLOCKS 1024
#define TPB    256
#define BUF_FLOATS (NCLS * TILE + TILE)   // 4864 real + 256 pad = 5120 floats
#define CHUNKS_PER_TILE 1280              // 5 * 256 lanes, padded from 1216

typedef float v2f __attribute__((ext_vector_type(2)));
typedef float v8f __attribute__((ext_vector_type(8)));

// ---------------------------------------------------------------------------
// Kernel 1: per-(h,w) any-over-batch 3-row label variation -> u8
// ---------------------------------------------------------------------------
__global__ void cl_rowvar_kernel(const int* __restrict__ tgt,
                                 unsigned char* __restrict__ rowvar) {
    int idx = blockIdx.x * blockDim.x + threadIdx.x;   // 0..HW-1
    int r = idx >> 9;                                   // W == 512
    unsigned char f = 0;
    if (r >= 1 && r <= H - 2) {
        for (int b = 0; b < BATCH; ++b) {
            int base = b * HW + idx;
            int v0 = tgt[base - W], v1 = tgt[base], v2 = tgt[base + W];
            if (!((v0 == v1) && (v1 == v2))) { f = 1; break; }
        }
    }
    rowvar[idx] = f;
}

// ---------------------------------------------------------------------------
// Kernel 2: horizontal 3-col dilation of rowvar -> boundary map u8
// ---------------------------------------------------------------------------
__global__ void cl_bmap_kernel(const unsigned char* __restrict__ rowvar,
                               unsigned char* __restrict__ bmap) {
    int idx = blockIdx.x * blockDim.x + threadIdx.x;
    int r = idx >> 9, c = idx & (W - 1);
    unsigned char v = 0;
    if (r >= 1 && r <= H - 2 && c >= 1 && c <= W - 2)
        v = (unsigned char)(rowvar[idx - 1] | rowvar[idx] | rowvar[idx + 1]);
    bmap[idx] = v;
}

// ---------------------------------------------------------------------------
// Kernel 3: main streaming pass. Async global->LDS double buffering.
// partials[block][64]: [0]=focal [1]=nll [2]=smooth [3]=bnll
//                      [4+c]=probsum [23+c]=inter [42+c]=count [61..63]=0
// ---------------------------------------------------------------------------
__device__ __forceinline__ void cl_issue_tile(const float* __restrict__ in,
                                              int tile, unsigned lds_base) {
    const int b  = tile >> 10;                 // HW/TILE = 1024 tiles per image
    const int s0 = (tile & 1023) << 8;
    const int t  = threadIdx.x;
#pragma unroll
    for (int i = 0; i < 5; ++i) {
        int idx  = i * TPB + t;                // 0..1279 (uniform lane activity)
        int cidx = idx >> 6;                   // 64 float4 chunks per class row
        int gc   = cidx > (NCLS - 1) ? (NCLS - 1) : cidx;   // pad reads row 18
        const float* gp = in + (size_t)(b * NCLS + gc) * HW
                             + (size_t)s0 + (size_t)((idx & 63) << 2);
        unsigned lva = lds_base + (unsigned)(idx << 4);     // 16B per chunk
        unsigned long long ga = (unsigned long long)(uintptr_t)gp;
        asm volatile("global_load_async_to_lds_b128 %0, %1, off"
                     :: "v"(lva), "v"(ga) : "memory");
    }
}

__global__ void __launch_bounds__(TPB)
cl_main_kernel(const float* __restrict__ in, const int* __restrict__ tgt,
               const unsigned char* __restrict__ bmap,
               float* __restrict__ partials) {
    __shared__ float smem[2 * BUF_FLOATS];
    __shared__ float red[64];

    const int t = threadIdx.x;
    const unsigned lds0 = (unsigned)(uintptr_t)(&smem[0]);

    float facc = 0.f, nacc = 0.f, sacc = 0.f, bacc = 0.f;
    float prob[NCLS], inter[NCLS], cnt[NCLS];
#pragma unroll
    for (int c = 0; c < NCLS; ++c) { prob[c] = 0.f; inter[c] = 0.f; cnt[c] = 0.f; }

    // prologue: stage first tile into buffer 0
    cl_issue_tile(in, blockIdx.x, lds0);

#pragma unroll 1
    for (int i = 0; i < NTILES / NBLOCKS; ++i) {
        const int tile = blockIdx.x + i * NBLOCKS;
        const int buf  = i & 1;
        const bool has_next = (i + 1) < (NTILES / NBLOCKS);
        if (has_next)
            cl_issue_tile(in, tile + NBLOCKS,
                          lds0 + (unsigned)((buf ^ 1) * BUF_FLOATS * 4));
        // wait for THIS wave's oldest 5 async loads (current tile) to land
        if (has_next) asm volatile("s_wait_asynccnt 0x5" ::: "memory");
        else          asm volatile("s_wait_asynccnt 0x0" ::: "memory");
        __syncthreads();   // all waves' portions of the tile are now in LDS

        const float* xs = smem + buf * BUF_FLOATS;
        const int b  = tile >> 10;
        const int s0 = (tile & 1023) << 8;
        const int pix = s0 + t;
        const int tv  = tgt[b * HW + pix];
        const float bmv = (float)bmap[pix];

        float x[NCLS];
        float m = -3.4e38f, xt = 0.f, sumx = 0.f;
#pragma unroll
        for (int c = 0; c < NCLS; ++c) {
            float v = xs[c * TILE + t];
            x[c] = v;
            m = fmaxf(m, v);
            sumx += v;
            xt = (c == tv) ? v : xt;
        }
        float se = 0.f;
#pragma unroll
        for (int c = 0; c < NCLS; ++c) { x[c] = __expf(x[c] - m); se += x[c]; }
        const float lse = m + __logf(se);
        const float inv = 1.f / se;
#pragma unroll
        for (int c = 0; c < NCLS; ++c) {
            float pc = x[c] * inv;
            prob[c] += pc;
            bool is = (c == tv);
            inter[c] += is ? pc : 0.f;
            cnt[c]   += is ? 1.f : 0.f;
        }
        const float nll = lse - xt;
        const float pt  = __expf(-nll);
        const float om  = 1.f - pt;
        facc += om * om * nll;
        nacc += nll;
        sacc += ((float)NCLS * lse - sumx) * (1.f / (float)NCLS);
        bacc += nll * bmv;

        __syncthreads();   // protect buf before it is re-filled next iter
    }

    // block reduction via LDS float atomics (ds_add_f32), once per block
    for (int i2 = t; i2 < 64; i2 += TPB) red[i2] = 0.f;
    __syncthreads();
    atomicAdd(&red[0], facc); atomicAdd(&red[1], nacc);
    atomicAdd(&red[2], sacc); atomicAdd(&red[3], bacc);
#pragma unroll
    for (int c = 0; c < NCLS; ++c) {
        atomicAdd(&red[4 + c],  prob[c]);
        atomicAdd(&red[23 + c], inter[c]);
        atomicAdd(&red[42 + c], cnt[c]);
    }
    __syncthreads();
    for (int i2 = t; i2 < 64; i2 += TPB)
        partials[(size_t)blockIdx.x * 64 + i2] = red[i2];
}

// ---------------------------------------------------------------------------
// Kernel 4: WMMA column-sum reduction of partials[1024][64] + final combine.
// D(16x16) = ones(16x4) x B(4x16) + C  => every D row holds column sums.
// B layout (f32 4x16, 2 VGPRs): v0 lanes0-15=K0, lanes16-31=K2; v1: K1 / K3.
// ---------------------------------------------------------------------------
__global__ void __launch_bounds__(128)
cl_finalize_kernel(const float* __restrict__ partials, float* __restrict__ out) {
    __shared__ float colsum[64];
    const int wave = threadIdx.x >> 5;       // 4 waves, 16 cols each
    const int lane = threadIdx.x & 31;
    const int c0   = wave * 16;
    const int half = lane >> 4;
    const int col  = c0 + (lane & 15);

    v8f acc = {};
    v2f a;  a[0] = 1.0f; a[1] = 1.0f;        // ones A-matrix in all lanes
#pragma unroll 1
    for (int r = 0; r < NBLOCKS; r += 4) {
        v2f bv;
        bv[0] = partials[(size_t)(r + half * 2 + 0) * 64 + col];
        bv[1] = partials[(size_t)(r + half * 2 + 1) * 64 + col];
        acc = __builtin_amdgcn_wmma_f32_16x16x4_f32(
                  false, a, false, bv, (short)0, acc, false, false);
    }
    if (lane < 16) colsum[c0 + lane] = acc[0];   // D row M=0: lanes 0-15, vgpr0
    __syncthreads();

    if (threadIdx.x == 0) {
        const float invN = 1.f / (float)NPIX;
        float focal    = colsum[0] * invN;
        float ce       = (0.9f * colsum[1] + 0.1f * colsum[2]) * invN;
        float boundary = (colsum[1] + 0.5f * colsum[3]) * invN;
        float ds = 0.f;
        for (int c = 0; c < NCLS; ++c) {
            float num = 2.f * colsum[23 + c] + 1e-5f;
            float den = colsum[4 + c] + colsum[42 + c] + 1e-5f;
            ds += 1.f - num / den;
        }
        float dice = ds / (float)NCLS;
        out[0] = focal; out[1] = dice; out[2] = ce; out[3] = boundary;
        out[4] = focal + dice + ce + boundary;
    }
}

// ---------------------------------------------------------------------------
extern "C" void kernel_launch(void* const* d_in, const int* in_sizes, int n_in,
                              void* d_out, int out_size, void* d_ws, size_t ws_size,
                              hipStream_t stream) {
    (void)in_sizes; (void)n_in; (void)out_size; (void)ws_size;
    const float* inputs = (const float*)d_in[0];
    const int*   targets = (const int*)d_in[1];
    float* out = (float*)d_out;

    unsigned char* rowvar = (unsigned char*)d_ws;                  // 256 KB
    unsigned char* bmap   = (unsigned char*)d_ws + HW;             // 256 KB
    float* partials       = (float*)((unsigned char*)d_ws + 2 * HW); // 256 KB

    cl_rowvar_kernel<<<HW / TPB, TPB, 0, stream>>>(targets, rowvar);
    cl_bmap_kernel<<<HW / TPB, TPB, 0, stream>>>(rowvar, bmap);
    cl_main_kernel<<<NBLOCKS, TPB, 0, stream>>>(inputs, targets, bmap, partials);
    cl_finalize_kernel<<<1, 128, 0, stream>>>(partials, out);
}